// DisplacementMap_14199161880932
// MI455X (gfx1250) — compile-verified
//
#include <hip/hip_runtime.h>

// dense_image_warp (tfa-style bilinear warp) for fixed shapes B=8,H=512,W=512,C=16.
// Memory-bound gather: ~285 MB of compulsory HBM traffic -> ~12 us floor at 23.3 TB/s.
// Strategy: 4 wave32 lanes per pixel, one float4 (b128) of channels per lane
// (adjacent lanes touch adjacent 16B -> fully coalesced b128s); non-temporal
// policy on the streaming flow/output so the 134 MB image can live in the
// 192 MB L2 (bilinear 4-neighbor reuse); gfx1250 WGP-scope global_prefetch on
// the flow stream.

typedef float f2 __attribute__((ext_vector_type(2)));
typedef float f4 __attribute__((ext_vector_type(4)));

namespace {
constexpr int kB = 8, kH = 512, kW = 512, kC = 16;
constexpr int kNPix = kB * kH * kW;       // 2^21 pixels
constexpr int kBlock = 256;               // 8 wave32 waves per workgroup
constexpr int kPixPerBlock = kBlock / 4;  // 64 pixels per block
}  // namespace

__device__ __forceinline__ f4 lerp4(f4 a, f4 b, float t) {
  f4 r;
  r.x = __builtin_fmaf(t, b.x - a.x, a.x);
  r.y = __builtin_fmaf(t, b.y - a.y, a.y);
  r.z = __builtin_fmaf(t, b.z - a.z, a.z);
  r.w = __builtin_fmaf(t, b.w - a.w, a.w);
  return r;
}

__global__ __launch_bounds__(kBlock) void dense_warp_kernel(
    const float* __restrict__ img, const f2* __restrict__ flow,
    f4* __restrict__ out) {
  const int gid = (int)blockIdx.x * kBlock + (int)threadIdx.x;
  const int pix = gid >> 2;  // pixel id in [0, 2^21)
  const int cg  = gid & 3;   // channel group: 4 floats each

  // WGP-scope prefetch of the next block's flow slab (gfx1250
  // global_prefetch_b8, fills all cache levels; wrapped to stay in-bounds
  // since WGP-scope prefetches are treated as non-speculative; no counter
  // cost).
  __builtin_prefetch(&flow[(pix + kPixPerBlock) & (kNPix - 1)], 0, 3);

  // flow is streamed exactly once -> non-temporal, keep L2 for the image.
  const f2 fl = __builtin_nontemporal_load(&flow[pix]);

  const int hw  = pix & (kH * kW - 1);
  const float py = (float)(hw >> 9);        // row   (W = 2^9)
  const float px = (float)(hw & (kW - 1));  // col
  const float qy = py - fl.x;               // query row
  const float qx = px - fl.y;               // query col

  // tfa interpolate_bilinear clipping semantics: floor clipped to [0, size-2],
  // alpha = clip(q - floor_clipped, 0, 1).
  float fy = __builtin_floorf(qy);
  float fx = __builtin_floorf(qx);
  fy = fminf(fmaxf(fy, 0.0f), (float)(kH - 2));
  fx = fminf(fmaxf(fx, 0.0f), (float)(kW - 2));
  const float ay = fminf(fmaxf(qy - fy, 0.0f), 1.0f);
  const float ax = fminf(fmaxf(qx - fx, 0.0f), 1.0f);
  const int iy = (int)fy;
  const int ix = (int)fx;
  const int b  = pix >> 18;  // H*W = 2^18

  // float index of this lane's 4-channel chunk at the top-left neighbor.
  const long long base =
      ((long long)((b * kH + iy) * kW + ix)) * kC + (cg << 2);
  const f4* p = (const f4*)(img + base);
  constexpr int kRight = kC / 4;         // +1 column, in f4 units
  constexpr int kDown  = (kW * kC) / 4;  // +1 row,    in f4 units

  // Four coalesced b128 gathers; default (RT) policy -> img tiles stay in L2.
  const f4 tl = p[0];
  const f4 tr = p[kRight];
  const f4 bl = p[kDown];
  const f4 br = p[kDown + kRight];

  const f4 top = lerp4(tl, tr, ax);
  const f4 bot = lerp4(bl, br, ax);
  const f4 res = lerp4(top, bot, ay);

  // Output is written exactly once -> non-temporal b128 store.
  __builtin_nontemporal_store(res, &out[(long long)pix * (kC / 4) + cg]);
}

extern "C" void kernel_launch(void* const* d_in, const int* in_sizes, int n_in,
                              void* d_out, int out_size, void* d_ws,
                              size_t ws_size, hipStream_t stream) {
  (void)in_sizes; (void)n_in; (void)out_size; (void)d_ws; (void)ws_size;
  const float* img = (const float*)d_in[0];
  const f2*    flw = (const f2*)d_in[1];
  f4*          out = (f4*)d_out;

  // 2^21 pixels * 4 lanes/pixel = 8,388,608 threads = 32768 blocks of 256 (exact).
  const int total_threads = kNPix * 4;
  dim3 grid(total_threads / kBlock), block(kBlock);
  hipLaunchKernelGGL(dense_warp_kernel, grid, block, 0, stream, img, flw, out);
}